// BatchHardTripletLoss_57612691309013
// MI455X (gfx1250) — compile-verified
//
#include <hip/hip_runtime.h>

// Batch-hard triplet loss, fused fp32-WMMA Gram + masked row reduction.
// B=4096, D=128. Data (2MB) is L2-resident on MI455X; the critical path is
// staging latency + the ~4.3 GFLOP fp32 WMMA Gram, so we double-buffer LDS
// panels and use gfx1250 async global->LDS DMA (ASYNCcnt-tracked).

#define MARGIN 0.3f
#define BIGF   1e30f

typedef __attribute__((ext_vector_type(2))) float v2f;
typedef __attribute__((ext_vector_type(8))) float v8f;

constexpr int D              = 128;
constexpr int ROWS_PER_BLOCK = 128;  // 8 waves * 16 rows
constexpr int COLS_PER_BLOCK = 512;  // per grid.y slice
constexpr int COL_CHUNK      = 64;   // columns staged in LDS at a time
constexpr int LDS_STRIDE     = 132;  // padded floats per column (132%4==0 for b128 stores,
                                     // 132%64==4 -> conflict-free tile reads)
constexpr int NCHUNK         = COLS_PER_BLOCK / COL_CHUNK;  // 8

#if defined(__has_builtin)
#if __has_builtin(__builtin_amdgcn_global_load_async_to_lds_b128) && \
    __has_builtin(__builtin_amdgcn_s_wait_asynccnt)
#define HAVE_ASYNC_LDS 1
#endif
#endif

#ifdef HAVE_ASYNC_LDS
// Builtin signature (from the round-2 diagnostic): param0 is
// `int __attribute__((vector_size(16))) __device__ *` (AS1), param1 mirrors in
// LDS (AS3). Launder generic pointers through integers (low 32 bits of a
// generic LDS pointer are the LDS byte offset per the aperture mapping).
typedef int v4i_b __attribute__((vector_size(4 * sizeof(int))));
#define AS1CAST(p) ((__attribute__((address_space(1))) v4i_b*)(unsigned long long)(const void*)(p))
#define AS3CAST(p) ((__attribute__((address_space(3))) v4i_b*)(unsigned int)(unsigned long long)(const void*)(p))
#endif

// ---------------- kernel 0: row norms + init hp/hn encodings ----------------
__global__ void bh_init(const float* __restrict__ E, float* __restrict__ sq,
                        unsigned* __restrict__ hp, unsigned* __restrict__ hn, int B) {
  int i = blockIdx.x * blockDim.x + threadIdx.x;
  if (i >= B) return;
  const float4* r = (const float4*)(E + (size_t)i * D);
  float s = 0.f;
#pragma unroll
  for (int j = 0; j < D / 4; ++j) {
    float4 v = r[j];
    s += v.x * v.x + v.y * v.y + v.z * v.z + v.w * v.w;
  }
  sq[i] = s;
  hp[i] = __float_as_uint(0.0f);  // distances >= 0 -> uint order == float order
  hn[i] = __float_as_uint(BIGF);
}

// ---- staging helpers: 64 cols x 128 dims (4 threads/col, 8 x b128 each) ----
#ifdef HAVE_ASYNC_LDS
__device__ inline void stage_async(const float* __restrict__ E, int colBase,
                                   float* buf, int tid) {
  int c = tid >> 2, part = tid & 3;
  const float* src = E + (size_t)(colBase + c) * D + part * 32;
  float* dst = &buf[c * LDS_STRIDE + part * 32];
#pragma unroll
  for (int j = 0; j < 8; ++j)
    __builtin_amdgcn_global_load_async_to_lds_b128(AS1CAST(src + j * 4),
                                                   AS3CAST(dst + j * 4), 0, 0);
}
#else
__device__ inline void load_regs(const float* __restrict__ E, int colBase,
                                 float4* tmp, int tid) {
  int c = tid >> 2, part = tid & 3;
  const float4* src = (const float4*)(E + (size_t)(colBase + c) * D) + part * 8;
#pragma unroll
  for (int j = 0; j < 8; ++j) tmp[j] = src[j];
}
__device__ inline void store_regs(float* buf, const float4* tmp, int tid) {
  int c = tid >> 2, part = tid & 3;
  float4* dst = (float4*)(&buf[c * LDS_STRIDE + part * 32]);
#pragma unroll
  for (int j = 0; j < 8; ++j) dst[j] = tmp[j];
}
#endif

// ---------------- kernel 1: fused WMMA Gram + masked hardest pos/neg --------
__global__ void __launch_bounds__(256)
bh_main(const float* __restrict__ E, const int* __restrict__ labels,
        const float* __restrict__ sq, unsigned* __restrict__ hp,
        unsigned* __restrict__ hn, int B) {
  __shared__ float ldsB[2][COL_CHUNK * LDS_STRIDE];  // double-buffered panels

  const int tid  = threadIdx.x;
  const int lane = tid & 31;
  const int wave = tid >> 5;
  const int rowBase  = blockIdx.x * ROWS_PER_BLOCK + wave * 16;
  const int colBlock = blockIdx.y * COLS_PER_BLOCK;

  const int hrow  = lane & 15;        // A: row-in-tile / B,C: N index
  const int khalf = (lane >> 4) * 2;  // K sub-offset for hi half-wave
  const int rsel  = (lane >> 4) * 8;  // C layout: hi half-wave holds M = r+8

  // --- A fragments: this wave's 16x128 strip, resident in VGPRs (32 x v2f) ---
  // ISA 16x4 fp32 A layout: lanes 0-15 hold K={0,1}, lanes 16-31 hold K={2,3}.
  const float* arow = E + (size_t)(rowBase + hrow) * D;
  v2f a[32];
#pragma unroll
  for (int kk = 0; kk < 32; ++kk)
    a[kk] = *(const v2f*)(arow + kk * 4 + khalf);

  // per-lane metadata for the rows this lane sees in the C layout
  float sqr[8];
  int   lr[8];
#pragma unroll
  for (int r = 0; r < 8; ++r) {
    int row = rowBase + r + rsel;
    sqr[r] = sq[row];
    lr[r]  = labels[row];
  }

  float hpl[8], hnl[8];  // running max/min for (row r, this lane's col slot)
#pragma unroll
  for (int r = 0; r < 8; ++r) { hpl[r] = -BIGF; hnl[r] = BIGF; }

  // ---- prologue: stage chunk 0 into buffer 0 ----
#ifdef HAVE_ASYNC_LDS
  stage_async(E, colBlock, ldsB[0], tid);
  __builtin_amdgcn_s_wait_asynccnt(0);
#else
  {
    float4 t[8];
    load_regs(E, colBlock, t, tid);
    store_regs(ldsB[0], t, tid);
  }
#endif
  __syncthreads();

  for (int chunk = 0; chunk < NCHUNK; ++chunk) {
    const int  cur     = chunk & 1;
    const bool hasNext = (chunk + 1) < NCHUNK;

    // ---- prefetch next panel while this one is consumed by WMMA ----
#ifdef HAVE_ASYNC_LDS
    if (hasNext) stage_async(E, colBlock + (chunk + 1) * COL_CHUNK, ldsB[cur ^ 1], tid);
#else
    float4 tmp[8];
    if (hasNext) load_regs(E, colBlock + (chunk + 1) * COL_CHUNK, tmp, tid);
#endif

    const int colBase = colBlock + chunk * COL_CHUNK;
    const float* buf  = ldsB[cur];

#pragma unroll
    for (int nt = 0; nt < COL_CHUNK / 16; ++nt) {
      v8f acc = {};
      // B 4x16 fp32 layout mirrors A: lanes 0-15 rows K={0,1}, lanes 16-31 K={2,3}.
      const float* bcol = &buf[(nt * 16 + hrow) * LDS_STRIDE + khalf];
#pragma unroll
      for (int kk = 0; kk < 32; ++kk) {
        v2f b = *(const v2f*)(bcol + kk * 4);
        acc = __builtin_amdgcn_wmma_f32_16x16x4_f32(
            /*neg_a=*/false, a[kk], /*neg_b=*/false, b,
            /*c_mod=*/(short)0, acc, /*reuse_a=*/false, /*reuse_b=*/false);
      }

      const int col = colBase + nt * 16 + hrow;
      const float sqc = sq[col];
      const int   lc  = labels[col];
#pragma unroll
      for (int r = 0; r < 8; ++r) {
        int row = rowBase + r + rsel;
        float d2 = fmaxf(sqr[r] + sqc - 2.0f * acc[r], 0.0f);
        float dist = sqrtf(d2);
        bool same = (lr[r] == lc);
        bool diag = (row == col);
        float pv = (same && !diag) ? dist : -BIGF;
        float nv = (!same) ? dist : BIGF;
        hpl[r] = fmaxf(hpl[r], pv);
        hnl[r] = fminf(hnl[r], nv);
      }
    }

    // ---- complete the prefetch, then publish the next buffer ----
#ifdef HAVE_ASYNC_LDS
    if (hasNext) __builtin_amdgcn_s_wait_asynccnt(0);
#else
    if (hasNext) store_regs(ldsB[cur ^ 1], tmp, tid);
#endif
    __syncthreads();
  }

  // fold across the 16 lanes of each half-wave (once, at the end)
#pragma unroll
  for (int r = 0; r < 8; ++r) {
#pragma unroll
    for (int m = 1; m < 16; m <<= 1) {
      hpl[r] = fmaxf(hpl[r], __shfl_xor(hpl[r], m, 16));
      hnl[r] = fminf(hnl[r], __shfl_xor(hnl[r], m, 16));
    }
  }
  if (hrow == 0) {
#pragma unroll
    for (int r = 0; r < 8; ++r) {
      int row = rowBase + r + rsel;
      if (hpl[r] > -0.5f * BIGF) atomicMax(&hp[row], __float_as_uint(hpl[r]));
      if (hnl[r] <  0.5f * BIGF) atomicMin(&hn[row], __float_as_uint(hnl[r]));
    }
  }
}

// ---------------- kernel 2: validity histogram + mean over valid rows -------
__global__ void bh_finish(const int* __restrict__ labels,
                          const unsigned* __restrict__ hp,
                          const unsigned* __restrict__ hn,
                          float* __restrict__ out, int B) {
  __shared__ int   hist[512];
  __shared__ float ssum[256];
  __shared__ float scnt[256];
  const int tid = threadIdx.x;

  for (int i = tid; i < 512; i += 256) hist[i] = 0;
  __syncthreads();
  for (int i = tid; i < B; i += 256) {
    int l = labels[i];
    if (l >= 0 && l < 512) atomicAdd(&hist[l], 1);
  }
  __syncthreads();

  float s = 0.f, c = 0.f;
  for (int i = tid; i < B; i += 256) {
    int l = labels[i];
    int cnt = (l >= 0 && l < 512) ? hist[l] : 1;
    if (cnt >= 2 && cnt < B) {  // has_pos && has_neg
      float v = __uint_as_float(hp[i]) - __uint_as_float(hn[i]) + MARGIN;
      s += fmaxf(v, 0.0f);
      c += 1.0f;
    }
  }
  ssum[tid] = s;
  scnt[tid] = c;
  __syncthreads();
  for (int off = 128; off > 0; off >>= 1) {
    if (tid < off) { ssum[tid] += ssum[tid + off]; scnt[tid] += scnt[tid + off]; }
    __syncthreads();
  }
  if (tid == 0) out[0] = (scnt[0] > 0.f) ? ssum[0] / scnt[0] : 0.0f;
}

// ---------------------------------------------------------------------------
extern "C" void kernel_launch(void* const* d_in, const int* in_sizes, int n_in,
                              void* d_out, int out_size, void* d_ws, size_t ws_size,
                              hipStream_t stream) {
  const float* E      = (const float*)d_in[0];
  const int*   labels = (const int*)d_in[1];
  const int B = in_sizes[1];  // 4096

  float*    sq = (float*)d_ws;
  unsigned* hp = (unsigned*)((char*)d_ws + (size_t)B * sizeof(float));
  unsigned* hn = hp + B;

  bh_init<<<(B + 255) / 256, 256, 0, stream>>>(E, sq, hp, hn, B);

  dim3 grid(B / ROWS_PER_BLOCK, B / COLS_PER_BLOCK);  // 32 x 8
  bh_main<<<grid, 256, 0, stream>>>(E, labels, sq, hp, hn, B);

  bh_finish<<<1, 256, 0, stream>>>(labels, hp, hn, (float*)d_out, B);
}